// BigSolDBPretrainModel_14439680049633
// MI455X (gfx1250) — compile-verified
//
#include <hip/hip_runtime.h>
#include <hip/hip_bf16.h>
#include <stdint.h>

typedef __attribute__((ext_vector_type(2))) float v2f;
typedef __attribute__((ext_vector_type(8))) float v8f;

#define HID 128
#define NGRAPH 256
#define CDIM 288   // 2*HID + 32

#if __has_builtin(__builtin_amdgcn_global_load_async_to_lds_b32) && \
    __has_builtin(__builtin_amdgcn_s_wait_asynccnt)
#define HAVE_ASYNC_LDS 1
typedef __attribute__((address_space(1))) int async_gptr_t;
typedef __attribute__((address_space(3))) int async_lptr_t;
#endif

// ---------------------------------------------------------------- utilities
__global__ void zero_kernel(float* __restrict__ p, long long n) {
    long long i = (long long)blockIdx.x * blockDim.x + threadIdx.x;
    if (i < n) p[i] = 0.0f;
}

// ---------------------------------------------------------------- embedding: x[N,9] @ w[9,128] + b
__global__ void embed_kernel(const float* __restrict__ x9, const float* __restrict__ w,
                             const float* __restrict__ b, float* __restrict__ xout, long long N) {
    long long node = blockIdx.x;
    int t = threadIdx.x;                 // 128 threads
    __shared__ float xs[9];
    if (t < 9) xs[t] = x9[node * 9 + t];
    __syncthreads();
    float s = b[t];
#pragma unroll
    for (int k = 0; k < 9; ++k) s = fmaf(xs[k], w[k * HID + t], s);
    xout[node * HID + t] = s;
}

// ---------------------------------------------------------------- agg init: agg = (1+eps)*x
__global__ void init_agg_kernel(const float* __restrict__ x, const float* __restrict__ eps,
                                float* __restrict__ agg, long long n) {
    long long i = (long long)blockIdx.x * blockDim.x + threadIdx.x;
    if (i < n) agg[i] = (1.0f + eps[0]) * x[i];
}

// ---------------------------------------------------------------- scatter: agg[dst] += x[src] (128 ch)
// One wave per edge: lanes share the (scalarized) index loads, float4 gather is a
// coalesced 512B wave transaction, scatter via global_atomic_add_f32.
__global__ void scatter_add_kernel(const float* __restrict__ x, const long long* __restrict__ ei,
                                   float* __restrict__ agg, long long E) {
    long long idx = (long long)blockIdx.x * blockDim.x + threadIdx.x;
    if (idx >= E * 32) return;           // 32 groups of 4 channels
    long long e = idx >> 5;
    int cg = (int)(idx & 31) << 2;
    long long s = ei[e];
    long long d = ei[E + e];
    const float4 v = *(const float4*)(x + s * HID + cg);
    float* p = agg + d * HID + cg;
    atomicAdd(p + 0, v.x);
    atomicAdd(p + 1, v.y);
    atomicAdd(p + 2, v.z);
    atomicAdd(p + 3, v.w);
}

// ---------------------------------------------------------------- fp32 WMMA GEMM
// Y = f(A)[M,K] @ W[K,NN] + bias;  f(a) = APPLY_BN ? relu(a*scale[k]+shift[k]) : a.
// Each wave owns TPW adjacent 16x16 N-tiles of one 16-row M-tile: the A operand
// (16x4 f32, 2 VGPRs; lane<16 = K{k,k+1}, lane>=16 = K{k+2,k+3}) is loaded once per
// k-step and reused by TPW back-to-back v_wmma_f32_16x16x4_f32 issues. 4 waves per
// block cover the full N extent (4*TPW*16 == NN). C/D per ISA layout: VGPR r ->
// M = r + 8*(lane/16), N = lane&15.
template <int TPW, bool APPLY_BN>
__global__ void wmma_gemm_kernel(const float* __restrict__ A, const float* __restrict__ W,
                                 const float* __restrict__ bias,
                                 const float* __restrict__ scale, const float* __restrict__ shift,
                                 float* __restrict__ Y, long long M, int K, int NN) {
    const int wave = threadIdx.x >> 5;
    const int lane = threadIdx.x & 31;
    const int l16  = lane & 15;
    const int hi   = lane >> 4;
    const long long m0 = (long long)blockIdx.x * 16;
    const int n0 = wave * (TPW * 16);

    long long arow = m0 + l16;
    if (arow >= M) arow = M - 1;
    const float* Ap = A + arow * K + 2 * hi;
    const float* Wp = W + (long long)(2 * hi) * NN + n0 + l16;

    v8f acc[TPW];
#pragma unroll
    for (int t = 0; t < TPW; ++t) acc[t] = (v8f){};

#pragma unroll 4
    for (int k = 0; k < K; k += 4) {
        float2 av = *(const float2*)(Ap + k);
        if (APPLY_BN) {
            int ka = k + 2 * hi;
            av.x = fmaxf(fmaf(av.x, scale[ka],     shift[ka]),     0.0f);
            av.y = fmaxf(fmaf(av.y, scale[ka + 1], shift[ka + 1]), 0.0f);
        }
        v2f a; a.x = av.x; a.y = av.y;
#pragma unroll
        for (int t = 0; t < TPW; ++t) {
            v2f b;
            b.x = Wp[(long long)k * NN + 16 * t];
            b.y = Wp[(long long)(k + 1) * NN + 16 * t];
            acc[t] = __builtin_amdgcn_wmma_f32_16x16x4_f32(false, a, false, b, (short)0,
                                                           acc[t], false, false);
        }
    }

#pragma unroll
    for (int t = 0; t < TPW; ++t) {
        const float bv = bias[n0 + 16 * t + l16];
#pragma unroll
        for (int r = 0; r < 8; ++r) {
            long long m = m0 + r + 8 * hi;
            if (m < M) Y[m * NN + n0 + 16 * t + l16] = acc[t][r] + bv;
        }
    }
}

// ---------------------------------------------------------------- per-column sum / sumsq partials
__global__ void colstats_partial_kernel(const float* __restrict__ y, float* __restrict__ sum,
                                        float* __restrict__ sq, long long M, int NN) {
    int c = threadIdx.x;                 // blockDim.x == NN (coalesced row reads)
    long long r0 = (long long)blockIdx.x * 64;
    long long r1 = r0 + 64; if (r1 > M) r1 = M;
    float s = 0.0f, ss = 0.0f;
    for (long long r = r0; r < r1; ++r) {
        float v = y[r * NN + c];
        s += v;
        ss = fmaf(v, v, ss);
    }
    atomicAdd(&sum[c], s);
    atomicAdd(&sq[c], ss);
}

__global__ void colstats_finalize_kernel(const float* __restrict__ sum, const float* __restrict__ sq,
                                         const float* __restrict__ g, const float* __restrict__ be,
                                         float* __restrict__ scale, float* __restrict__ shift,
                                         float invN) {
    int c = threadIdx.x + blockIdx.x * blockDim.x;
    float m   = sum[c] * invN;
    float var = fmaf(-m, m, sq[c] * invN);
    float sc  = g[c] * rsqrtf(var + 1e-5f);
    scale[c] = sc;
    shift[c] = fmaf(-m, sc, be[c]);
}

// ---------------------------------------------------------------- BN + ReLU in place (C = 128)
__global__ void bn_relu_kernel(float* __restrict__ h, const float* __restrict__ scale,
                               const float* __restrict__ shift, long long n) {
    long long i = (long long)blockIdx.x * blockDim.x + threadIdx.x;
    if (i >= n) return;
    int c = (int)(i & (HID - 1));
    h[i] = fmaxf(fmaf(h[i], scale[c], shift[c]), 0.0f);
}

// ---------------------------------------------------------------- mean-pool into concat buffer
__global__ void pool_kernel(const float* __restrict__ x, const long long* __restrict__ batch,
                            float* __restrict__ cbuf, float* __restrict__ cnt,
                            long long N, int base) {
    long long idx = (long long)blockIdx.x * blockDim.x + threadIdx.x;
    if (idx >= N * HID) return;
    long long node = idx >> 7;
    int c = (int)(idx & (HID - 1));
    long long gph = batch[node];
    atomicAdd(&cbuf[gph * CDIM + base + c], x[idx]);
    if (c == 0) atomicAdd(&cnt[gph], 1.0f);
}

__global__ void pool_div_kernel(float* __restrict__ cbuf, const float* __restrict__ cnt, int base) {
    int gph = blockIdx.x;                // 256 graphs
    int c = threadIdx.x;                 // 128
    cbuf[gph * CDIM + base + c] /= fmaxf(cnt[gph], 1.0f);
}

// ---------------------------------------------------------------- temperature MLP: [256,1]->32->32
__global__ void temp_mlp_kernel(const float* __restrict__ t_in, const float* __restrict__ w1,
                                const float* __restrict__ b1, const float* __restrict__ w2,
                                const float* __restrict__ b2, float* __restrict__ cbuf) {
    int gph = threadIdx.x + blockIdx.x * blockDim.x;
    if (gph >= NGRAPH) return;
    float tv = t_in[gph];
    float h[32];
#pragma unroll
    for (int j = 0; j < 32; ++j) h[j] = fmaxf(fmaf(tv, w1[j], b1[j]), 0.0f);
    for (int j2 = 0; j2 < 32; ++j2) {
        float s = b2[j2];
#pragma unroll
        for (int j = 0; j < 32; ++j) s = fmaf(h[j], w2[j * 32 + j2], s);
        cbuf[gph * CDIM + 2 * HID + j2] = s;
    }
}

// ---------------------------------------------------------------- predictor head: 288->128->64->1
__global__ void pred_mlp_kernel(const float* __restrict__ cbuf,
                                const float* __restrict__ w1, const float* __restrict__ b1,
                                const float* __restrict__ w2, const float* __restrict__ b2,
                                const float* __restrict__ w3, const float* __restrict__ b3,
                                float* __restrict__ out) {
    __shared__ float cs[CDIM];
    __shared__ float h1[128];
    __shared__ float h2[64];
    int gph = blockIdx.x;
    int t = threadIdx.x;                 // 128
#if defined(HAVE_ASYNC_LDS)
    // Stage the concat row through the gfx1250 async global->LDS path.
    for (int i = t; i < CDIM; i += 128)
        __builtin_amdgcn_global_load_async_to_lds_b32(
            (async_gptr_t*)(cbuf + (long long)gph * CDIM + i),
            (async_lptr_t*)&cs[i], 0, 0);
    __builtin_amdgcn_s_wait_asynccnt(0);
    __syncthreads();
#else
    for (int i = t; i < CDIM; i += 128) cs[i] = cbuf[(long long)gph * CDIM + i];
    __syncthreads();
#endif
    float s = b1[t];
    for (int k = 0; k < CDIM; ++k) s = fmaf(cs[k], w1[k * 128 + t], s);
    h1[t] = fmaxf(s, 0.0f);
    __syncthreads();
    if (t < 64) {
        float s2 = b2[t];
#pragma unroll 4
        for (int k = 0; k < 128; ++k) s2 = fmaf(h1[k], w2[k * 64 + t], s2);
        h2[t] = fmaxf(s2, 0.0f) * w3[t];
    }
    __syncthreads();
    if (t == 0) {
        float o = b3[0];
#pragma unroll
        for (int k = 0; k < 64; ++k) o += h2[k];
        out[gph] = o;
    }
}

// ================================================================ host orchestration
static inline void zero_launch(float* p, long long n, hipStream_t s) {
    zero_kernel<<<(unsigned)((n + 255) / 256), 256, 0, s>>>(p, n);
}

struct EncPtr {
    const float* x9;
    const long long* ei;
    const long long* batch;
    int emb_b, emb_w, layer0;   // d_in indices; layer L leaves at layer0 + 9*L
};

static void run_encoder(void* const* d_in, const EncPtr& ep, long long N, long long E,
                        float* xbuf, float* aggbuf, float* ybuf,
                        float* sum0, float* sq0, float* scl, float* shf,
                        float* cnt, float* cbuf, int base, hipStream_t stream) {
    const long long nh = N * HID;
    const unsigned bNH = (unsigned)((nh + 255) / 256);
    const unsigned mt = (unsigned)((N + 15) / 16);

    embed_kernel<<<(unsigned)N, HID, 0, stream>>>(
        ep.x9, (const float*)d_in[ep.emb_w], (const float*)d_in[ep.emb_b], xbuf, N);

    for (int L = 0; L < 4; ++L) {
        const int lb = ep.layer0 + 9 * L;   // b1,b2,be1,be2,eps,g1,g2,w1,w2
        const float* b1  = (const float*)d_in[lb + 0];
        const float* b2  = (const float*)d_in[lb + 1];
        const float* be1 = (const float*)d_in[lb + 2];
        const float* be2 = (const float*)d_in[lb + 3];
        const float* eps = (const float*)d_in[lb + 4];
        const float* g1  = (const float*)d_in[lb + 5];
        const float* g2  = (const float*)d_in[lb + 6];
        const float* w1  = (const float*)d_in[lb + 7];
        const float* w2  = (const float*)d_in[lb + 8];

        init_agg_kernel<<<bNH, 256, 0, stream>>>(xbuf, eps, aggbuf, nh);
        scatter_add_kernel<<<(unsigned)((E * 32 + 255) / 256), 256, 0, stream>>>(
            xbuf, ep.ei, aggbuf, E);

        // GEMM1: y[N,256] = agg @ w1 + b1  (4 waves x 4 tiles = 256 cols)
        wmma_gemm_kernel<4, false><<<mt, 128, 0, stream>>>(
            aggbuf, w1, b1, nullptr, nullptr, ybuf, N, HID, 2 * HID);

        zero_launch(sum0, 256, stream);
        zero_launch(sq0, 256, stream);
        colstats_partial_kernel<<<(unsigned)((N + 63) / 64), 2 * HID, 0, stream>>>(
            ybuf, sum0, sq0, N, 2 * HID);
        colstats_finalize_kernel<<<1, 2 * HID, 0, stream>>>(
            sum0, sq0, g1, be1, scl, shf, 1.0f / (float)N);

        // GEMM2 with BN+ReLU fused into A-load: x[N,128] = relu(bn(y)) @ w2 + b2
        wmma_gemm_kernel<2, true><<<mt, 128, 0, stream>>>(
            ybuf, w2, b2, scl, shf, xbuf, N, 2 * HID, HID);

        zero_launch(sum0, HID, stream);
        zero_launch(sq0, HID, stream);
        colstats_partial_kernel<<<(unsigned)((N + 63) / 64), HID, 0, stream>>>(
            xbuf, sum0, sq0, N, HID);
        colstats_finalize_kernel<<<1, HID, 0, stream>>>(
            sum0, sq0, g2, be2, scl, shf, 1.0f / (float)N);

        bn_relu_kernel<<<bNH, 256, 0, stream>>>(xbuf, scl, shf, nh);
    }

    zero_launch(cnt, NGRAPH, stream);
    pool_kernel<<<bNH, 256, 0, stream>>>(xbuf, ep.batch, cbuf, cnt, N, base);
    pool_div_kernel<<<NGRAPH, HID, 0, stream>>>(cbuf, cnt, base);
}

extern "C" void kernel_launch(void* const* d_in, const int* in_sizes, int n_in,
                              void* d_out, int out_size, void* d_ws, size_t ws_size,
                              hipStream_t stream) {
    (void)n_in; (void)out_size; (void)ws_size;
    const long long N = in_sizes[0] / 9;
    const long long E = in_sizes[1] / 2;

    // params pytree (sorted-key flatten) starts at index 7:
    //   drug:    emb_b=7, emb_w=8, layers 9..44 (9 leaves x 4)
    //   pred:    b1=45, b2=46, b3=47, w1=48, w2=49, w3=50
    //   solvent: emb_b=51, emb_w=52, layers 53..88
    //   temp:    b1=89, b2=90, w1=91, w2=92
    EncPtr drug = { (const float*)d_in[0], (const long long*)d_in[1],
                    (const long long*)d_in[2], 7, 8, 9 };
    EncPtr solv = { (const float*)d_in[3], (const long long*)d_in[4],
                    (const long long*)d_in[5], 51, 52, 53 };

    float* ws = (float*)d_ws;
    float* xbuf   = ws;                        // N*128
    float* aggbuf = xbuf + N * HID;            // N*128
    float* ybuf   = aggbuf + N * HID;          // N*256
    float* sum0   = ybuf + N * 2 * HID;        // 256
    float* sq0    = sum0 + 256;                // 256
    float* scl    = sq0 + 256;                 // 256
    float* shf    = scl + 256;                 // 256
    float* cnt    = shf + 256;                 // 256
    float* cbuf   = cnt + 256;                 // 256*288 concat [drug|solv|temp]

    zero_launch(cbuf, (long long)NGRAPH * CDIM, stream);

    run_encoder(d_in, drug, N, E, xbuf, aggbuf, ybuf, sum0, sq0, scl, shf, cnt, cbuf, 0, stream);
    run_encoder(d_in, solv, N, E, xbuf, aggbuf, ybuf, sum0, sq0, scl, shf, cnt, cbuf, HID, stream);

    temp_mlp_kernel<<<1, NGRAPH, 0, stream>>>(
        (const float*)d_in[6], (const float*)d_in[91], (const float*)d_in[89],
        (const float*)d_in[92], (const float*)d_in[90], cbuf);

    pred_mlp_kernel<<<NGRAPH, 128, 0, stream>>>(
        cbuf,
        (const float*)d_in[48], (const float*)d_in[45],
        (const float*)d_in[49], (const float*)d_in[46],
        (const float*)d_in[50], (const float*)d_in[47],
        (float*)d_out);
}